// SeerMambaTransLayer2_50637664420398
// MI455X (gfx1250) — compile-verified
//
#include <hip/hip_runtime.h>
#include <math.h>
#include <stdint.h>

// ---------------- problem dims ----------------
#define CH    128
#define BATCH 2
#define SEQ   512
#define BL    (BATCH * SEQ)    // 1024 tokens
#define DMM   256              // mamba d_model
#define DI    512              // d_inner
#define DS    128              // d_state
#define DTR   16               // dt_rank
#define RTOT  (DTR + 2 * DS)   // 272

typedef __attribute__((ext_vector_type(2))) float v2f;
typedef __attribute__((ext_vector_type(4))) float v4f;
typedef __attribute__((ext_vector_type(8))) float v8f;

// -------- optional CDNA5 async global->LDS path (compile-probed) --------
#if defined(__has_builtin)
#if __has_builtin(__builtin_amdgcn_global_load_async_to_lds_b128)
#define HAVE_ASYNC_LDS 1
#endif
#endif

#if defined(HAVE_ASYNC_LDS)
// Signature (from hipcc diagnostic): param0 is a non-const pointer to
// 'int __attribute__((vector_size(16)))' in AS1 (printed '__device__').
typedef int v4i_vs __attribute__((vector_size(4 * sizeof(int))));
typedef __attribute__((address_space(1))) v4i_vs gv4i;
typedef __attribute__((address_space(3))) v4i_vs lv4i;

__device__ __forceinline__ void async_cp16(const float* g, float* l) {
  // low 32 bits of a generic LDS pointer are the wave-relative LDS offset
  __builtin_amdgcn_global_load_async_to_lds_b128(
      (gv4i*)(uintptr_t)g,
      (lv4i*)(uint32_t)(uintptr_t)l,
      0, 0);
}
#if __has_builtin(__builtin_amdgcn_s_wait_asynccnt)
#define WAIT_ASYNC(n) do { __builtin_amdgcn_s_wait_asynccnt(n); \
                           asm volatile("" ::: "memory"); } while (0)
#else
#define WAIT_ASYNC(n) asm volatile("s_wait_asynccnt %0" :: "i"(n) : "memory")
#endif
#endif

// =====================================================================
// f32 GEMM on V_WMMA_F32_16X16X4_F32.   C[M,N] = A[M,K] * B[N,K]^T
// (B always stored row=n, contiguous in k; ldb = k-stride)
// One wave owns one 16x16 tile.  K must be a multiple of 8.
// Per 8-wide K chunk: lanes 0-15 take global k {k0..k0+3} (HW K-slots 0,1),
// lanes 16-31 take {k0+4..k0+7} (HW K-slots 2,3); one b128 load per operand
// feeds two WMMAs (.xy then .zw) — K-sum order is permutation-invariant.
// C/D layout: lane L -> N = L&15 ; VGPR v -> M = v + (L>=16 ? 8 : 0)
// =====================================================================
__global__ void gemm_wmma_f32(const float* __restrict__ A, int lda,
                              const float* __restrict__ B, int ldb,
                              float* __restrict__ C, int ldc,
                              int M, int N, int K,
                              const float* __restrict__ bias,
                              const float* __restrict__ res, int ldres,
                              int act)   // 0 = none, 1 = softplus
{
  const int wid  = threadIdx.x >> 5;
  const int lane = threadIdx.x & 31;
  const int nTN  = (N + 15) >> 4;
  const int nTM  = (M + 15) >> 4;
  const int t    = blockIdx.x * (blockDim.x >> 5) + wid;
  if (t >= nTM * nTN) return;                  // wave-uniform
  const int tm = t / nTN, tn = t % nTN;

  const int r    = lane & 15;
  const int koff = (lane >> 4) << 2;           // 0 (lanes 0-15) or 4 (16-31)
  const int arow = min(tm * 16 + r, M - 1);
  const int bcol = min(tn * 16 + r, N - 1);

  const float* __restrict__ ap = A + (size_t)arow * lda + koff;
  const float* __restrict__ bp = B + (size_t)bcol * ldb + koff;

  v8f acc = {};
#pragma unroll 2
  for (int k0 = 0; k0 < K; k0 += 8) {
    v4f av = *(const v4f*)(ap + k0);
    v4f bv = *(const v4f*)(bp + k0);
    if (k0 + 40 <= K) {                         // uniform guard
      __builtin_prefetch(ap + k0 + 32, 0, 3);   // -> global_prefetch_b8
      __builtin_prefetch(bp + k0 + 32, 0, 3);
    }
    v2f a0 = {av.x, av.y}, a1 = {av.z, av.w};
    v2f b0 = {bv.x, bv.y}, b1 = {bv.z, bv.w};
    acc = __builtin_amdgcn_wmma_f32_16x16x4_f32(false, a0, false, b0,
                                                (short)0, acc, false, false);
    acc = __builtin_amdgcn_wmma_f32_16x16x4_f32(false, a1, false, b1,
                                                (short)0, acc, false, false);
  }

  const int col = tn * 16 + r;
  if (col < N) {
    const int rbase = tm * 16 + ((lane >> 4) << 3);
    const float bv = bias ? bias[col] : 0.f;
#pragma unroll
    for (int v = 0; v < 8; ++v) {
      int row = rbase + v;
      if (row < M) {
        float val = acc[v] + bv;
        if (act == 1) val = (val > 20.f) ? val : log1pf(expf(val));  // softplus
        if (res) val += res[(size_t)row * ldres + col];
        C[(size_t)row * ldc + col] = val;
      }
    }
  }
}

// ---------------- small transpose: out[c*R + r] = in[r*Cc + c] ----------------
__global__ void transpose_k(const float* __restrict__ in, float* __restrict__ out,
                            int R, int Cc) {
  int idx = blockIdx.x * blockDim.x + threadIdx.x;
  if (idx >= R * Cc) return;
  int r = idx / Cc, c = idx % Cc;
  out[(size_t)c * R + r] = in[(size_t)r * Cc + c];
}

// ---------------- attention: sim = (Q . K) / sqrt(C) ----------------
__global__ void sim_kernel(const float* __restrict__ q,
                           const float* __restrict__ Kbuf,
                           float* __restrict__ sim) {
  int idx = blockIdx.x * blockDim.x + threadIdx.x;   // b*8*SEQ
  if (idx >= BATCH * 8 * SEQ) return;
  int l = idx % SEQ, n = (idx / SEQ) % 8, b = idx / (SEQ * 8);
  const float* qp = q + b * CH + n * 16;
  const float* kp = Kbuf + ((size_t)(b * SEQ + l)) * CH + n * 16;
  float s = 0.f;
#pragma unroll
  for (int h = 0; h < 16; ++h) s += qp[h] * kp[h];
  sim[idx] = s * 0.08838834764831845f;               // 1/sqrt(128)
}

// ---------------- softmax over l per (b, head); 512 threads ----------------
__global__ void softmax_kernel(const float* __restrict__ sim,
                               float* __restrict__ Aw) {
  const float* s = sim + (size_t)blockIdx.x * SEQ;
  int wid = threadIdx.x >> 5, lane = threadIdx.x & 31;
  __shared__ float red[16];
  float v = s[threadIdx.x];

  float m = v;
  for (int off = 16; off > 0; off >>= 1) m = fmaxf(m, __shfl_xor(m, off, 32));
  if (lane == 0) red[wid] = m;
  __syncthreads();
  if (wid == 0) {
    float mm = (lane < 16) ? red[lane] : -INFINITY;
    for (int off = 8; off > 0; off >>= 1) mm = fmaxf(mm, __shfl_xor(mm, off, 32));
    if (lane == 0) red[0] = mm;
  }
  __syncthreads();
  m = red[0];
  float e = expf(v - m);
  float ss = e;
  for (int off = 16; off > 0; off >>= 1) ss += __shfl_xor(ss, off, 32);
  __syncthreads();
  if (lane == 0) red[wid] = ss;
  __syncthreads();
  if (wid == 0) {
    float tt = (lane < 16) ? red[lane] : 0.f;
    for (int off = 8; off > 0; off >>= 1) tt += __shfl_xor(tt, off, 32);
    if (lane == 0) red[0] = tt;
  }
  __syncthreads();
  Aw[(size_t)blockIdx.x * SEQ + threadIdx.x] = e / red[0];
}

// ---------------- attn reduce + CLS2 = CLS + attn ----------------
__global__ void attn_kernel(const float* __restrict__ Aw,
                            const float* __restrict__ Vbuf,
                            const float* __restrict__ CLS,
                            float* __restrict__ CLS2ws,
                            float* __restrict__ CLS2out) {
  int idx = blockIdx.x * blockDim.x + threadIdx.x;   // BATCH*CH = 256
  if (idx >= BATCH * CH) return;
  int c = idx % CH, b = idx / CH;
  const float* aw = Aw + (size_t)(b * 8 + (c >> 4)) * SEQ;
  const float* vp = Vbuf + (size_t)b * SEQ * CH + c;
  float s = 0.f;
  for (int l = 0; l < SEQ; ++l) s += aw[l] * vp[(size_t)l * CH];
  float o = CLS[idx] + s;
  CLS2ws[idx]  = o;
  CLS2out[idx] = o;
}

// ---------------- LN1 over concat([x, CLS2]) (256 features) ----------------
__global__ void ln1_kernel(const float* __restrict__ x,
                           const float* __restrict__ CLS2,
                           const float* __restrict__ w,
                           const float* __restrict__ bb,
                           float* __restrict__ hn) {
  int token = blockIdx.x, b = token / SEQ, f = threadIdx.x;
  float v = (f < CH) ? x[(size_t)token * CH + f] : CLS2[b * CH + (f - CH)];
  __shared__ float red[8];
  __shared__ float stats[2];
  int wid = f >> 5, lane = f & 31;

  float s = v;
  for (int off = 16; off > 0; off >>= 1) s += __shfl_xor(s, off, 32);
  if (lane == 0) red[wid] = s;
  __syncthreads();
  if (f == 0) { float t = 0; for (int i = 0; i < 8; ++i) t += red[i]; stats[0] = t * (1.f / 256.f); }
  __syncthreads();
  float d = v - stats[0];
  float sq = d * d;
  for (int off = 16; off > 0; off >>= 1) sq += __shfl_xor(sq, off, 32);
  if (lane == 0) red[wid] = sq;
  __syncthreads();
  if (f == 0) { float t = 0; for (int i = 0; i < 8; ++i) t += red[i]; stats[1] = rsqrtf(t * (1.f / 256.f) + 1e-5f); }
  __syncthreads();
  hn[(size_t)token * DMM + f] = d * stats[1] * w[f] + bb[f];
}

// ---------------- LN2 over 128 features ----------------
__global__ void ln2_kernel(const float* __restrict__ xin,
                           const float* __restrict__ w,
                           const float* __restrict__ bb,
                           float* __restrict__ outn) {
  int token = blockIdx.x, f = threadIdx.x;
  float v = xin[(size_t)token * CH + f];
  __shared__ float red[4];
  __shared__ float stats[2];
  int wid = f >> 5, lane = f & 31;

  float s = v;
  for (int off = 16; off > 0; off >>= 1) s += __shfl_xor(s, off, 32);
  if (lane == 0) red[wid] = s;
  __syncthreads();
  if (f == 0) { stats[0] = (red[0] + red[1] + red[2] + red[3]) * (1.f / 128.f); }
  __syncthreads();
  float d = v - stats[0];
  float sq = d * d;
  for (int off = 16; off > 0; off >>= 1) sq += __shfl_xor(sq, off, 32);
  if (lane == 0) red[wid] = sq;
  __syncthreads();
  if (f == 0) { stats[1] = rsqrtf((red[0] + red[1] + red[2] + red[3]) * (1.f / 128.f) + 1e-5f); }
  __syncthreads();
  outn[(size_t)token * CH + f] = d * stats[1] * w[f] + bb[f];
}

// ---------------- depthwise causal conv(4) + SiLU ----------------
__global__ void conv_silu_kernel(const float* __restrict__ xz,
                                 const float* __restrict__ cw,
                                 const float* __restrict__ cb,
                                 float* __restrict__ xc_act) {
  int idx = blockIdx.x * blockDim.x + threadIdx.x;
  if (idx >= BL * DI) return;
  int d = idx % DI, token = idx / DI, l = token % SEQ, b = token / SEQ;
  float s = cb[d];
#pragma unroll
  for (int j = 0; j < 4; ++j) {
    int ls = l - 3 + j;
    if (ls >= 0) s += cw[d * 4 + j] * xz[((size_t)(b * SEQ + ls)) * (2 * DI) + d];
  }
  xc_act[idx] = s / (1.f + expf(-s));                 // SiLU
}

// ---------------- selective scan: 1 wave per (b, d), 4 states/lane ----------
// B/C rows (256 f32 / token) double-buffered into LDS via async DMA when
// the toolchain exposes global_load_async_to_lds_b128.
__global__ void scan_gate_kernel(const float* __restrict__ delta,
                                 const float* __restrict__ dbl,
                                 const float* __restrict__ xc_act,
                                 const float* __restrict__ xz,
                                 const float* __restrict__ A_log,
                                 const float* __restrict__ Dp,
                                 float* __restrict__ ygate) {
  int wid = threadIdx.x >> 5, lane = threadIdx.x & 31;
  int row = blockIdx.x * 4 + wid;                     // 0..BATCH*DI-1
  int b = row >> 9, d = row & (DI - 1);
  int nb = lane * 4;

  float Ar[4], h[4];
#pragma unroll
  for (int i = 0; i < 4; ++i) { Ar[i] = -expf(A_log[d * DS + nb + i]); h[i] = 0.f; }
  const float Dd = Dp[d];

#if defined(HAVE_ASYNC_LDS)
  __shared__ float sbuf[4][2][256];                   // per-wave double buffer
  float* buf0 = &sbuf[wid][0][0];
  float* buf1 = &sbuf[wid][1][0];
  {
    const float* g = dbl + (size_t)(b * SEQ) * RTOT + DTR;
    async_cp16(g + nb,       buf0 + nb);              // B block (128 f32)
    async_cp16(g + 128 + nb, buf0 + 128 + nb);        // C block (128 f32)
  }
#endif

  for (int l = 0; l < SEQ; ++l) {
    int token = b * SEQ + l;
    float dl  = delta [(size_t)token * DI + d];
    float xcv = xc_act[(size_t)token * DI + d];

    float Bv[4], Cv[4];
#if defined(HAVE_ASYNC_LDS)
    float* cur = (l & 1) ? buf1 : buf0;
    if (l + 1 < SEQ) {
      float* nxt = (l & 1) ? buf0 : buf1;
      const float* g = dbl + (size_t)(token + 1) * RTOT + DTR;
      async_cp16(g + nb,       nxt + nb);
      async_cp16(g + 128 + nb, nxt + 128 + nb);
      WAIT_ASYNC(2);                                  // current buffer landed
    } else {
      WAIT_ASYNC(0);
    }
    v4f bq = *(const v4f*)(cur + nb);
    v4f cq = *(const v4f*)(cur + 128 + nb);
    Bv[0] = bq.x; Bv[1] = bq.y; Bv[2] = bq.z; Bv[3] = bq.w;
    Cv[0] = cq.x; Cv[1] = cq.y; Cv[2] = cq.z; Cv[3] = cq.w;
#else
    const float* br = dbl + (size_t)token * RTOT;
    v4f bq = *(const v4f*)(br + DTR + nb);
    v4f cq = *(const v4f*)(br + DTR + DS + nb);
    Bv[0] = bq.x; Bv[1] = bq.y; Bv[2] = bq.z; Bv[3] = bq.w;
    Cv[0] = cq.x; Cv[1] = cq.y; Cv[2] = cq.z; Cv[3] = cq.w;
#endif

    float dx = dl * xcv;
    float acc = 0.f;
#pragma unroll
    for (int i = 0; i < 4; ++i) {
      float dA = expf(dl * Ar[i]);
      h[i] = dA * h[i] + dx * Bv[i];
      acc += h[i] * Cv[i];
    }
    for (int off = 16; off > 0; off >>= 1) acc += __shfl_xor(acc, off, 32);
    if (lane == 0) {
      float y = acc + xcv * Dd;
      float z = xz[(size_t)token * (2 * DI) + DI + d];
      y *= z / (1.f + expf(-z));                      // gate: silu(z)
      ygate[(size_t)token * DI + d] = y;
    }
  }
}

// =====================================================================
extern "C" void kernel_launch(void* const* d_in, const int* in_sizes, int n_in,
                              void* d_out, int out_size, void* d_ws, size_t ws_size,
                              hipStream_t stream) {
  const float* x     = (const float*)d_in[0];
  const float* CLS   = (const float*)d_in[1];
  const float* W_Q   = (const float*)d_in[2];
  const float* W_K   = (const float*)d_in[3];
  const float* W_V   = (const float*)d_in[4];
  const float* n1w   = (const float*)d_in[5];
  const float* n1b   = (const float*)d_in[6];
  const float* n2w   = (const float*)d_in[7];
  const float* n2b   = (const float*)d_in[8];
  const float* headw = (const float*)d_in[9];
  const float* headb = (const float*)d_in[10];
  const float* inw   = (const float*)d_in[11];
  const float* convw = (const float*)d_in[12];
  const float* convb = (const float*)d_in[13];
  const float* xpw   = (const float*)d_in[14];
  const float* dtw   = (const float*)d_in[15];
  const float* dtb   = (const float*)d_in[16];
  const float* Alog  = (const float*)d_in[17];
  const float* Dp    = (const float*)d_in[18];
  const float* opw   = (const float*)d_in[19];

  // workspace carve-up (floats)
  float* ws     = (float*)d_ws;
  float* Kbuf   = ws;                     // BL*CH
  float* Vbuf   = Kbuf   + BL * CH;       // BL*CH
  float* qbuf   = Vbuf   + BL * CH;       // BATCH*CH
  float* CLS2ws = qbuf   + BATCH * CH;    // BATCH*CH
  float* simb   = CLS2ws + BATCH * CH;    // BATCH*8*SEQ
  float* hn     = simb   + BATCH * 8 * SEQ;   // BL*DMM
  float* xzb    = hn     + BL * DMM;      // BL*2*DI
  float* xc_act = xzb    + BL * 2 * DI;   // BL*DI
  float* dblb   = xc_act + BL * DI;       // BL*RTOT
  float* deltab = dblb   + BL * RTOT;     // BL*DI
  float* ygate  = deltab + BL * DI;       // BL*DI
  float* xo1    = ygate  + BL * DI;       // BL*CH
  float* ln2b   = xo1    + BL * CH;       // BL*CH
  float* WQt    = ln2b   + BL * CH;       // CH*CH   (transposed weights)
  float* WKt    = WQt    + CH * CH;       // CH*CH
  float* WVt    = WKt    + CH * CH;       // CH*CH

  float* out      = (float*)d_out;
  float* out_xo   = out;                  // BL*CH
  float* out_CLS2 = out + BL * CH;        // BATCH*CH
  float* out_Aw   = out_CLS2 + BATCH * CH;// BATCH*8*SEQ

  auto gemm = [&](const float* A, int lda, const float* B, int ldb,
                  float* Cc, int ldc, int M, int N, int K,
                  const float* bias, const float* res, int ldres, int act) {
    int tiles  = ((M + 15) / 16) * ((N + 15) / 16);
    int blocks = (tiles + 3) / 4;                     // 4 waves / block
    hipLaunchKernelGGL(gemm_wmma_f32, dim3(blocks), dim3(128), 0, stream,
                       A, lda, B, ldb, Cc, ldc, M, N, K, bias, res, ldres, act);
  };

  // ---- transpose attention weights so all GEMMs use the B[n][k] fast path ----
  transpose_k<<<(CH * CH + 255) / 256, 256, 0, stream>>>(W_Q, WQt, CH, CH);
  transpose_k<<<(CH * CH + 255) / 256, 256, 0, stream>>>(W_K, WKt, CH, CH);
  transpose_k<<<(CH * CH + 255) / 256, 256, 0, stream>>>(W_V, WVt, CH, CH);

  // ---- attention ----
  gemm(x,   CH, WKt, CH, Kbuf, CH, BL,    CH, CH, nullptr, nullptr, 0, 0);
  gemm(x,   CH, WVt, CH, Vbuf, CH, BL,    CH, CH, nullptr, nullptr, 0, 0);
  gemm(CLS, CH, WQt, CH, qbuf, CH, BATCH, CH, CH, nullptr, nullptr, 0, 0);
  sim_kernel<<<(BATCH * 8 * SEQ + 255) / 256, 256, 0, stream>>>(qbuf, Kbuf, simb);
  softmax_kernel<<<BATCH * 8, SEQ, 0, stream>>>(simb, out_Aw);
  attn_kernel<<<1, BATCH * CH, 0, stream>>>(out_Aw, Vbuf, CLS, CLS2ws, out_CLS2);

  // ---- concat + LN1 ----
  ln1_kernel<<<BL, 256, 0, stream>>>(x, CLS2ws, n1w, n1b, hn);

  // ---- mamba ----
  gemm(hn, DMM, inw, DMM, xzb, 2 * DI, BL, 2 * DI, DMM, nullptr, nullptr, 0, 0);
  conv_silu_kernel<<<(BL * DI + 255) / 256, 256, 0, stream>>>(xzb, convw, convb, xc_act);
  gemm(xc_act, DI, xpw, DI, dblb, RTOT, BL, RTOT, DI, nullptr, nullptr, 0, 0);
  gemm(dblb, RTOT, dtw, DTR, deltab, DI, BL, DI, DTR, dtb, nullptr, 0, /*softplus*/1);
  scan_gate_kernel<<<(BATCH * DI) / 4, 128, 0, stream>>>(deltab, dblb, xc_act, xzb,
                                                         Alog, Dp, ygate);
  gemm(ygate, DI, opw, DI, xo1, CH, BL, CH, DI, nullptr, x, CH, 0);

  // ---- LN2 + head + residual ----
  ln2_kernel<<<BL, CH, 0, stream>>>(xo1, n2w, n2b, ln2b);
  gemm(ln2b, CH, headw, CH, out_xo, CH, BL, CH, CH, headb, xo1, CH, 0);
}